// Encoder_88983132439258
// MI455X (gfx1250) — compile-verified
//
#include <hip/hip_runtime.h>
#include <hip/hip_bf16.h>

// ---- problem constants (match reference) ----
#define N_GRAPHS   16
#define NODES_PER  4096
#define NNODES     (N_GRAPHS * NODES_PER)   // 65536
#define DEG        32
#define NEDGES     (NNODES * DEG)           // 2097152
#define FEAT_IN    16
#define HID        32
#define FC_HID     256
#define LATENT     64
#define KFLAT      (NODES_PER * HID)        // 131072
#define KCHUNK     (KFLAT / 128)            // 1024

typedef __attribute__((ext_vector_type(2))) float v2f;
typedef __attribute__((ext_vector_type(8))) float v8f;

__device__ __forceinline__ float elu_f(float v) {
    return v > 0.0f ? v : (__expf(v) - 1.0f);
}

// -------------------------------------------------------------------------
// Edge scatter, C=32: one wave per edge. Edge index/weight are wave-uniform
// -> scalarize via readfirstlane so src/dst/w come in on the scalar (SMEM)
// path; each lane handles one channel (coalesced 128B gather + 128B atomic).
// -------------------------------------------------------------------------
__global__ void scatter32_wave(const float* __restrict__ hin,
                               const long long* __restrict__ src,
                               const long long* __restrict__ dst,
                               const float* __restrict__ w,
                               float* __restrict__ agg) {
    int e = __builtin_amdgcn_readfirstlane(
        (int)((blockIdx.x * blockDim.x + threadIdx.x) >> 5));
    int lane = threadIdx.x & 31;
    int s = (int)src[e];
    int d = (int)dst[e];
    float we = w[e];
    atomicAdd(&agg[d * HID + lane], hin[s * HID + lane] * we);
}

// -------------------------------------------------------------------------
// Edge scatter, C=16: one wave per 2 edges (lanes 0-15 -> e0, 16-31 -> e1).
// Both edges' metadata loaded scalarly; per-half select.
// -------------------------------------------------------------------------
__global__ void scatter16_wave(const float* __restrict__ hin,
                               const long long* __restrict__ src,
                               const long long* __restrict__ dst,
                               const float* __restrict__ w,
                               float* __restrict__ agg) {
    int e0 = __builtin_amdgcn_readfirstlane(
        (int)((blockIdx.x * blockDim.x + threadIdx.x) >> 5)) * 2;
    int lane = threadIdx.x & 31;
    int half = lane >> 4;
    int c    = lane & 15;
    int s0 = (int)src[e0], s1 = (int)src[e0 + 1];
    int d0 = (int)dst[e0], d1 = (int)dst[e0 + 1];
    float w0 = w[e0], w1 = w[e0 + 1];
    int s = half ? s1 : s0;
    int d = half ? d1 : d0;
    float we = half ? w1 : w0;
    atomicAdd(&agg[d * FEAT_IN + c], hin[s * FEAT_IN + c] * we);
}

// -------------------------------------------------------------------------
// Per-node linear + ELU via f32 WMMA 16x16x4:
//   hout[N x 32] = ELU(agg[N x CIN] @ W[CIN x 32] + b)
// One wave per 16x16 output tile. Grid: (N/16)*2 waves, 8 waves/block.
// A 16x4 layout: lanes 0-15 K={0,1}, lanes 16-31 K={2,3}; B mirrored;
// C/D: VGPR r -> row r (lanes 0-15) / row r+8 (lanes 16-31).
// -------------------------------------------------------------------------
template <int CIN>
__global__ void node_linear_elu(const float* __restrict__ agg,
                                const float* __restrict__ W,
                                const float* __restrict__ bias,
                                float* __restrict__ hout) {
    int wave  = (blockIdx.x * blockDim.x + threadIdx.x) >> 5;
    int lane  = threadIdx.x & 31;
    int ntile = wave & 1;          // column tile (0..1)
    int mtile = wave >> 1;         // node tile (0..4095)
    int m0 = mtile * 16;
    int n0 = ntile * 16;
    int half = lane >> 4;
    int lm   = lane & 15;

    const float* arow = agg + (m0 + lm) * CIN;
    const float* bcol = W + n0 + lm;

    v2f a[CIN / 4], b[CIN / 4];
#pragma unroll
    for (int u = 0; u < CIN / 4; ++u) {
        int ka = u * 4 + half * 2;
        a[u] = *(const v2f*)(arow + ka);
        b[u].x = bcol[ka * HID];
        b[u].y = bcol[(ka + 1) * HID];
    }
    v8f acc = {};
#pragma unroll
    for (int u = 0; u < CIN / 4; ++u) {
        acc = __builtin_amdgcn_wmma_f32_16x16x4_f32(
            false, a[u], false, b[u], (short)0, acc, false, false);
    }
    float bv = bias[n0 + lm];
#pragma unroll
    for (int r = 0; r < 8; ++r) {
        int m = m0 + r + half * 8;
        hout[m * HID + n0 + lm] = elu_f(acc[r] + bv);
    }
}

// -------------------------------------------------------------------------
// FC1 partial GEMM via f32 WMMA: acc[16 x 256] += h[16 x 131072] @ Wfc1 tile.
// Grid: (16 n-tiles, 16 k-groups); block 256 = 8 waves; wave -> k-chunk of
// 1024 (scalarized bound -> uniform s_cbranch loop, EXEC stays all-ones).
// K unrolled x4 per iteration so 12 loads are in flight before the WMMAs.
// Each Wfc1 element read exactly once from HBM; h re-read 16x from L2.
// -------------------------------------------------------------------------
__global__ void fc1_wmma(const float* __restrict__ h,
                         const float* __restrict__ Wfc1,
                         float* __restrict__ acc_out) {
    int lane = threadIdx.x & 31;
    int n0   = blockIdx.x * 16;
    int kchunk = __builtin_amdgcn_readfirstlane(
        (int)(blockIdx.y * 8 + (threadIdx.x >> 5)));    // 0..127, wave-uniform
    int kbeg = kchunk * KCHUNK;
    int half = lane >> 4;
    int lm   = lane & 15;

    v8f acc = {};
    const float* arow = h + lm * KFLAT;                 // graph row m = lm
    const float* bcol = Wfc1 + n0 + lm;                 // column n = n0+lm

#pragma unroll 1
    for (int k0 = kbeg; k0 < kbeg + KCHUNK; k0 += 16) {
        v2f a[4], b[4];
#pragma unroll
        for (int u = 0; u < 4; ++u) {
            int ka = k0 + u * 4 + half * 2;
            a[u] = *(const v2f*)(arow + ka);            // b64 vector load
            b[u].x = bcol[(long long)ka * FC_HID];
            b[u].y = bcol[(long long)(ka + 1) * FC_HID];
        }
#pragma unroll
        for (int u = 0; u < 4; ++u) {
            acc = __builtin_amdgcn_wmma_f32_16x16x4_f32(
                false, a[u], false, b[u], (short)0, acc, false, false);
        }
    }
#pragma unroll
    for (int r = 0; r < 8; ++r) {
        int m = r + half * 8;
        atomicAdd(&acc_out[m * FC_HID + n0 + lm], acc[r]);
    }
}

// fc1h = ELU(fc1acc + bfc1)   (16*256 = 4096 elements)
__global__ void fc1_finish(const float* __restrict__ acc,
                           const float* __restrict__ bias,
                           float* __restrict__ out) {
    int t = blockIdx.x * blockDim.x + threadIdx.x;
    if (t < N_GRAPHS * FC_HID) {
        out[t] = elu_f(acc[t] + bias[t & (FC_HID - 1)]);
    }
}

// -------------------------------------------------------------------------
// FC2: out[16 x 64] = fc1h[16 x 256] @ Wfc2[256 x 64] + bfc2.
// One block, 4 waves (one per 16-wide N tile), K=256 -> 64 WMMA steps.
// -------------------------------------------------------------------------
__global__ void fc2_wmma(const float* __restrict__ fc1h,
                         const float* __restrict__ Wfc2,
                         const float* __restrict__ bfc2,
                         float* __restrict__ out) {
    int wave = threadIdx.x >> 5;
    int lane = threadIdx.x & 31;
    int n0 = wave * 16;
    int half = lane >> 4;
    int lm   = lane & 15;

    v8f acc = {};
    const float* arow = fc1h + lm * FC_HID;
    const float* bcol = Wfc2 + n0 + lm;
#pragma unroll 1
    for (int k0 = 0; k0 < FC_HID; k0 += 16) {
        v2f a[4], b[4];
#pragma unroll
        for (int u = 0; u < 4; ++u) {
            int ka = k0 + u * 4 + half * 2;
            a[u] = *(const v2f*)(arow + ka);
            b[u].x = bcol[ka * LATENT];
            b[u].y = bcol[(ka + 1) * LATENT];
        }
#pragma unroll
        for (int u = 0; u < 4; ++u) {
            acc = __builtin_amdgcn_wmma_f32_16x16x4_f32(
                false, a[u], false, b[u], (short)0, acc, false, false);
        }
    }
    float bv = bfc2[n0 + lm];
#pragma unroll
    for (int r = 0; r < 8; ++r) {
        int m = r + half * 8;
        out[m * LATENT + n0 + lm] = acc[r] + bv;
    }
}

// -------------------------------------------------------------------------
extern "C" void kernel_launch(void* const* d_in, const int* in_sizes, int n_in,
                              void* d_out, int out_size, void* d_ws, size_t ws_size,
                              hipStream_t stream) {
    const float* x    = (const float*)d_in[0];   // [N,16]
    const float* ea   = (const float*)d_in[1];   // [E]
    const float* W1   = (const float*)d_in[2];
    const float* b1   = (const float*)d_in[3];
    const float* W2   = (const float*)d_in[4];
    const float* b2   = (const float*)d_in[5];
    const float* W3   = (const float*)d_in[6];
    const float* b3   = (const float*)d_in[7];
    const float* Wfc1 = (const float*)d_in[8];   // [131072,256]
    const float* bfc1 = (const float*)d_in[9];
    const float* Wfc2 = (const float*)d_in[10];  // [256,64]
    const float* bfc2 = (const float*)d_in[11];
    const long long* eidx = (const long long*)d_in[12];  // [2,E] int64
    const long long* src = eidx;
    const long long* dst = eidx + NEDGES;

    // workspace layout (f32)
    float* agg    = (float*)d_ws;                    // N*32
    float* hA     = agg + (size_t)NNODES * HID;      // N*32
    float* hB     = hA  + (size_t)NNODES * HID;      // N*32
    float* fc1acc = hB  + (size_t)NNODES * HID;      // 16*256
    float* fc1h   = fc1acc + N_GRAPHS * FC_HID;      // 16*256

    const int linBlocks = (NNODES / 16) * 2 * 32 / 256;      // 1024
    const int sc32Blocks = NEDGES * 32 / 256;                // one wave per edge
    const int sc16Blocks = (NEDGES / 2) * 32 / 256;          // one wave per 2 edges

    // ---- layer 1: x (C=16) -> hA ----
    hipMemsetAsync(agg, 0, (size_t)NNODES * FEAT_IN * sizeof(float), stream);
    scatter16_wave<<<sc16Blocks, 256, 0, stream>>>(x, src, dst, ea, agg);
    node_linear_elu<FEAT_IN><<<linBlocks, 256, 0, stream>>>(agg, W1, b1, hA);

    // ---- layer 2: hA (C=32) -> hB ----
    hipMemsetAsync(agg, 0, (size_t)NNODES * HID * sizeof(float), stream);
    scatter32_wave<<<sc32Blocks, 256, 0, stream>>>(hA, src, dst, ea, agg);
    node_linear_elu<HID><<<linBlocks, 256, 0, stream>>>(agg, W2, b2, hB);

    // ---- layer 3: hB (C=32) -> hA ----
    hipMemsetAsync(agg, 0, (size_t)NNODES * HID * sizeof(float), stream);
    scatter32_wave<<<sc32Blocks, 256, 0, stream>>>(hB, src, dst, ea, agg);
    node_linear_elu<HID><<<linBlocks, 256, 0, stream>>>(agg, W3, b3, hA);

    // ---- FC1: [16 x 131072] @ [131072 x 256], K split across 128 waves ----
    hipMemsetAsync(fc1acc, 0, N_GRAPHS * FC_HID * sizeof(float), stream);
    fc1_wmma<<<dim3(FC_HID / 16, 16), 256, 0, stream>>>(hA, Wfc1, fc1acc);
    fc1_finish<<<(N_GRAPHS * FC_HID + 255) / 256, 256, 0, stream>>>(fc1acc, bfc1, fc1h);

    // ---- FC2: [16 x 256] @ [256 x 64] + b -> d_out ----
    fc2_wmma<<<1, 128, 0, stream>>>(fc1h, Wfc2, bfc2, (float*)d_out);
}